// The_fine_tune_network_46428596470444
// MI455X (gfx1250) — compile-verified
//
#include <hip/hip_runtime.h>
#include <hip/hip_bf16.h>
#include <math.h>

#define BATCH 8192
#define DIM 128
#define EPSF 1e-6f
#define MARGINF 0.1f
#define NEG_INF -3.402823466e+38f
#define KPAD 288   // 257 padded up to a multiple of 32 (chunk size)

typedef __attribute__((ext_vector_type(2))) float v2f;
typedef __attribute__((ext_vector_type(4))) float v4f;
typedef __attribute__((ext_vector_type(8))) float v8f;

// ---------------------------------------------------------------------------
// c[j] = -sum_k op[j][k]^2 + 2*EPS*sum_k op[j][k]   (per-column argmax const)
// ---------------------------------------------------------------------------
__global__ void col_const_kernel(const float* __restrict__ x, float* __restrict__ c) {
  int j = blockIdx.x * blockDim.x + threadIdx.x;
  if (j >= BATCH) return;
  const v4f* row = (const v4f*)(x + (size_t)(2 * j + 1) * DIM);
  float nb = 0.f, sb = 0.f;
#pragma unroll
  for (int k = 0; k < DIM / 4; ++k) {
    v4f v = row[k];
    nb += v.x * v.x + v.y * v.y + v.z * v.z + v.w * v.w;
    sb += v.x + v.y + v.z + v.w;
  }
  c[j] = -nb + 2.f * EPSF * sb;
}

// ---------------------------------------------------------------------------
// Fused distance-GEMM + row-wise argmax (hard negative mining).
// Block = 8 waves, owns 16 rows; wave w scans column tiles jt = w, w+8, ...
// A (16 rows x K=128) preloaded in 64 VGPRs; B double-buffered in chunks of
// 8 k-steps, pipelined across chunk AND tile boundaries so global_load_b64
// overlaps v_wmma_f32_16x16x4_f32.
// score(i,j) = 2*dot(oa_i, op_j) + c[j], diag masked to -inf.
// ---------------------------------------------------------------------------
__global__ __launch_bounds__(256) void argmax_kernel(const float* __restrict__ x,
                                                     const float* __restrict__ c,
                                                     int* __restrict__ negidx) {
  __shared__ float s_val[16][8];
  __shared__ int   s_idx[16][8];
  const int lane = threadIdx.x & 31;
  const int wave = threadIdx.x >> 5;
  const int r = lane & 15;       // row-in-tile for A, col-in-tile for B/C
  const int h = lane >> 4;       // half-wave selector
  const int m0 = blockIdx.x * 16;

  // Preload A fragments: rows m0..m0+15, K = 0..127 (32 k-steps of 4)
  v2f a[32];
  const float* oaRow = x + (size_t)(2 * (m0 + r)) * DIM + 2 * h;
#pragma unroll
  for (int s = 0; s < 32; ++s) a[s] = *(const v2f*)(oaRow + 4 * s);

  float bval[8];
  int   bidx[8];
#pragma unroll
  for (int v = 0; v < 8; ++v) { bval[v] = NEG_INF; bidx[v] = 0; }

  v2f bc[8], bn[8];
  // prologue: chunk 0 of first tile
  {
    const float* p0 = x + (size_t)(2 * (wave * 16 + r) + 1) * DIM + 2 * h;
#pragma unroll
    for (int s = 0; s < 8; ++s) bc[s] = *(const v2f*)(p0 + 4 * s);
  }

  for (int jt = wave; jt < BATCH / 16; jt += 8) {
    const float* rowC = x + (size_t)(2 * (jt * 16 + r) + 1) * DIM + 2 * h;
    const float* rowN = x + (size_t)(2 * ((jt + 8) * 16 + r) + 1) * DIM + 2 * h;
    const bool hasNext = (jt + 8) < (BATCH / 16);
    v8f acc = {0.f, 0.f, 0.f, 0.f, 0.f, 0.f, 0.f, 0.f};
#pragma unroll
    for (int ch = 0; ch < 4; ++ch) {
      // issue loads for the next chunk (next tile's chunk 0 when ch == 3)
      const float* nb = (ch < 3) ? (rowC + (ch + 1) * 32) : rowN;
      if (ch < 3 || hasNext) {
#pragma unroll
        for (int s = 0; s < 8; ++s) bn[s] = *(const v2f*)(nb + 4 * s);
      }
#pragma unroll
      for (int s = 0; s < 8; ++s)
        acc = __builtin_amdgcn_wmma_f32_16x16x4_f32(false, a[8 * ch + s], false, bc[s],
                                                    (short)0, acc, false, false);
#pragma unroll
      for (int s = 0; s < 8; ++s) bc[s] = bn[s];
    }
    const int col = jt * 16 + r;
    const float cj = c[col];
#pragma unroll
    for (int v = 0; v < 8; ++v) {
      const int row = m0 + v + 8 * h;
      float score = 2.f * acc[v] + cj;
      if (row == col) score = NEG_INF;
      if (score > bval[v]) { bval[v] = score; bidx[v] = col; }
    }
  }

  // Reduce over the 16 lanes (columns) holding the same row.
#pragma unroll
  for (int v = 0; v < 8; ++v) {
    float bv = bval[v];
    int   bi = bidx[v];
#pragma unroll
    for (int off = 8; off > 0; off >>= 1) {
      float ov = __shfl_xor(bv, off, 16);
      int   oi = __shfl_xor(bi, off, 16);
      if (ov > bv || (ov == bv && oi < bi)) { bv = ov; bi = oi; }
    }
    if (r == 0) { s_val[v + 8 * h][wave] = bv; s_idx[v + 8 * h][wave] = bi; }
  }
  __syncthreads();

  // Combine the 8 waves' candidates per row.
  if (threadIdx.x < 16) {
    float bv = NEG_INF;
    int   bi = 0;
    for (int w = 0; w < 8; ++w) {
      float ov = s_val[threadIdx.x][w];
      int   oi = s_idx[threadIdx.x][w];
      if (ov > bv || (ov == bv && oi < bi)) { bv = ov; bi = oi; }
    }
    negidx[m0 + threadIdx.x] = bi;
  }
}

// ---------------------------------------------------------------------------
// Build the 16384 x 288 MLP input (rows 0..B-1 = pos, rows B..2B-1 = neg).
// Layout per row: [score, ra(128), r{p|n}(128), zeros(31)]
// ---------------------------------------------------------------------------
__global__ __launch_bounds__(128) void build_inputs_kernel(const float* __restrict__ x,
                                                           const float* __restrict__ ori,
                                                           const int* __restrict__ negidx,
                                                           float* __restrict__ X0) {
  __shared__ float sp[128], sn[128];
  const int i = blockIdx.x;
  const int t = threadIdx.x;
  const int ni = negidx[i];
  float oa = x[(size_t)(2 * i) * DIM + t];
  float op = x[(size_t)(2 * i + 1) * DIM + t];
  float on = x[(size_t)(2 * ni + 1) * DIM + t];
  float dp = oa - op + EPSF;
  float dn = oa - on + EPSF;
  sp[t] = dp * dp;
  sn[t] = dn * dn;
  __syncthreads();
  for (int s2 = 64; s2 > 0; s2 >>= 1) {
    if (t < s2) { sp[t] += sp[t + s2]; sn[t] += sn[t + s2]; }
    __syncthreads();
  }
  float ra = ori[(size_t)(2 * i) * DIM + t];
  float rp = ori[(size_t)(2 * i + 1) * DIM + t];
  float rn = ori[(size_t)(2 * ni + 1) * DIM + t];
  float* posRow = X0 + (size_t)i * KPAD;
  float* negRow = X0 + (size_t)(BATCH + i) * KPAD;
  if (t == 0) { posRow[0] = sp[0]; negRow[0] = sn[0]; }
  posRow[1 + t] = ra;  posRow[129 + t] = rp;
  negRow[1 + t] = ra;  negRow[129 + t] = rn;
  if (t < KPAD - 257) { posRow[257 + t] = 0.f; negRow[257 + t] = 0.f; }
}

// ---------------------------------------------------------------------------
// Repack W (Ksrc x N, row-major) into WMMA fragment-major layout, zero-padding
// K up to Kdst. Output group g = (ntile * (Kdst/4) + kb) holds 64 floats:
// [lane 0..31][elem 0..1] with  k = kb*4 + 2*(lane>>4) + elem,  n = ntile*16 + (lane&15).
// Each lane in the GEMM then loads one contiguous float2 per k-step.
// ---------------------------------------------------------------------------
__global__ void pack_w_kernel(const float* __restrict__ W, float* __restrict__ Wp,
                              int Ksrc, int Kdst, int N) {
  int idx = blockIdx.x * blockDim.x + threadIdx.x;
  if (idx >= Kdst * N) return;
  int v    = idx & 1;
  int lane = (idx >> 1) & 31;
  int g    = idx >> 6;
  int kq   = Kdst >> 2;
  int kb    = g % kq;
  int ntile = g / kq;
  int k = kb * 4 + 2 * (lane >> 4) + v;
  int n = ntile * 16 + (lane & 15);
  Wp[idx] = (k < Ksrc) ? W[(size_t)k * N + n] : 0.f;
}

// ---------------------------------------------------------------------------
// f32 WMMA GEMM: Y[16384 x N] = act(X[16384 x K] @ W[K x N] + bias)
// W is pre-packed fragment-major. act: 0 = SiLU, 1 = sigmoid.
// Block = 8 waves -> 16 rows x 128 cols; K chunked by 32 with double-buffered
// register pipelines for both A and B fragments.
// ---------------------------------------------------------------------------
__global__ __launch_bounds__(256) void gemm_kernel(const float* __restrict__ X, int ldx,
                                                   const float* __restrict__ Wp,
                                                   const float* __restrict__ bias,
                                                   float* __restrict__ Y, int ldy,
                                                   int N, int K, int act) {
  const int lane = threadIdx.x & 31;
  const int wave = threadIdx.x >> 5;
  const int r = lane & 15;
  const int h = lane >> 4;
  const int n0 = blockIdx.x * 128 + wave * 16;
  if (n0 >= N) return;  // wave-uniform: EXEC stays all-ones for WMMA
  const int m0 = blockIdx.y * 16;

  const float* xRow  = X + (size_t)(m0 + r) * ldx + 2 * h;
  const float* wTile = Wp + ((size_t)(n0 >> 4) * (K >> 2)) * 64 + lane * 2;
  const int nc = K >> 5;  // chunks of 32 k-values (8 WMMA steps)

  v2f ac[8], bc[8], an[8], bn[8];
#pragma unroll
  for (int s = 0; s < 8; ++s) {
    ac[s] = *(const v2f*)(xRow + 4 * s);
    bc[s] = *(const v2f*)(wTile + s * 64);
  }

  v8f acc = {0.f, 0.f, 0.f, 0.f, 0.f, 0.f, 0.f, 0.f};
  for (int c2 = 0; c2 < nc; ++c2) {
    const bool more = (c2 + 1) < nc;
    if (more) {
      const float* xn = xRow + (c2 + 1) * 32;
      const float* wn = wTile + (size_t)(c2 + 1) * 8 * 64;
#pragma unroll
      for (int s = 0; s < 8; ++s) {
        an[s] = *(const v2f*)(xn + 4 * s);
        bn[s] = *(const v2f*)(wn + s * 64);
      }
    }
#pragma unroll
    for (int s = 0; s < 8; ++s)
      acc = __builtin_amdgcn_wmma_f32_16x16x4_f32(false, ac[s], false, bc[s],
                                                  (short)0, acc, false, false);
    if (more) {
#pragma unroll
      for (int s = 0; s < 8; ++s) { ac[s] = an[s]; bc[s] = bn[s]; }
    }
  }

  const int col = n0 + r;
  const float bv = bias[col];
#pragma unroll
  for (int v = 0; v < 8; ++v) {
    float y = acc[v] + bv;
    if (act == 0) y = y / (1.f + __expf(-y));        // SiLU
    else          y = 1.f / (1.f + __expf(-y));      // sigmoid
    Y[(size_t)(m0 + v + 8 * h) * ldy + col] = y;
  }
}

// Layer 6: s[i] = sigmoid(H5[i,:64] . W6 + b6), i in [0, 2B)
__global__ void final_layer_kernel(const float* __restrict__ H,
                                   const float* __restrict__ W6,
                                   const float* __restrict__ b6,
                                   float* __restrict__ s) {
  int i = blockIdx.x * blockDim.x + threadIdx.x;
  if (i >= 2 * BATCH) return;
  const float* row = H + (size_t)i * 64;
  float acc = b6[0];
#pragma unroll
  for (int k = 0; k < 64; ++k) acc += row[k] * W6[k];
  s[i] = 1.f / (1.f + __expf(-acc));
}

// loss = mean_i relu(ps_i^2 - ns_i^2 + margin)
__global__ __launch_bounds__(1024) void loss_kernel(const float* __restrict__ s,
                                                    float* __restrict__ out) {
  __shared__ float red[1024];
  int t = threadIdx.x;
  float acc = 0.f;
  for (int i = t; i < BATCH; i += 1024) {
    float ps = s[i], ns = s[BATCH + i];
    float v = ps * ps - ns * ns + MARGINF;
    acc += (v > 0.f) ? v : 0.f;
  }
  red[t] = acc;
  __syncthreads();
  for (int s2 = 512; s2 > 0; s2 >>= 1) {
    if (t < s2) red[t] += red[t + s2];
    __syncthreads();
  }
  if (t == 0) out[0] = red[0] / (float)BATCH;
}

// ---------------------------------------------------------------------------
extern "C" void kernel_launch(void* const* d_in, const int* in_sizes, int n_in,
                              void* d_out, int out_size, void* d_ws, size_t ws_size,
                              hipStream_t stream) {
  const float* x   = (const float*)d_in[0];
  const float* ori = (const float*)d_in[1];
  const float* W1 = (const float*)d_in[2];  const float* b1 = (const float*)d_in[3];
  const float* W2 = (const float*)d_in[4];  const float* b2 = (const float*)d_in[5];
  const float* W3 = (const float*)d_in[6];  const float* b3 = (const float*)d_in[7];
  const float* W4 = (const float*)d_in[8];  const float* b4 = (const float*)d_in[9];
  const float* W5 = (const float*)d_in[10]; const float* b5 = (const float*)d_in[11];
  const float* W6 = (const float*)d_in[12]; const float* b6 = (const float*)d_in[13];

  // Workspace layout (floats)
  float* bufA = (float*)d_ws;                       // 16384*512 (also holds X0 16384x288)
  float* bufB = bufA + (size_t)16384 * 512;         // 16384*512
  float* ccol = bufB + (size_t)16384 * 512;         // 8192
  int*   nidx = (int*)(ccol + BATCH);               // 8192
  float* sres = (float*)(nidx + BATCH);             // 16384
  float* Wp1  = sres + 2 * BATCH;                   // 288*512
  float* Wp2  = Wp1 + (size_t)KPAD * 512;           // 512*512
  float* Wp3  = Wp2 + (size_t)512 * 512;            // 512*256
  float* Wp4  = Wp3 + (size_t)512 * 256;            // 256*128
  float* Wp5  = Wp4 + (size_t)256 * 128;            // 128*64

  col_const_kernel<<<BATCH / 256, 256, 0, stream>>>(x, ccol);
  pack_w_kernel<<<(KPAD * 512 + 255) / 256, 256, 0, stream>>>(W1, Wp1, 257, KPAD, 512);
  pack_w_kernel<<<(512 * 512 + 255) / 256, 256, 0, stream>>>(W2, Wp2, 512, 512, 512);
  pack_w_kernel<<<(512 * 256 + 255) / 256, 256, 0, stream>>>(W3, Wp3, 512, 512, 256);
  pack_w_kernel<<<(256 * 128 + 255) / 256, 256, 0, stream>>>(W4, Wp4, 256, 256, 128);
  pack_w_kernel<<<(128 * 64 + 255) / 256, 256, 0, stream>>>(W5, Wp5, 128, 128, 64);

  argmax_kernel<<<BATCH / 16, 256, 0, stream>>>(x, ccol, nidx);
  build_inputs_kernel<<<BATCH, 128, 0, stream>>>(x, ori, nidx, bufA);

  // MLP: 16384-row batch; grid.y = 16384/16 = 1024
  gemm_kernel<<<dim3(4, 1024), 256, 0, stream>>>(bufA, KPAD, Wp1, b1, bufB, 512, 512, KPAD, 0);
  gemm_kernel<<<dim3(4, 1024), 256, 0, stream>>>(bufB, 512, Wp2, b2, bufA, 512, 512, 512, 0);
  gemm_kernel<<<dim3(2, 1024), 256, 0, stream>>>(bufA, 512, Wp3, b3, bufB, 256, 256, 512, 0);
  gemm_kernel<<<dim3(1, 1024), 256, 0, stream>>>(bufB, 256, Wp4, b4, bufA, 128, 128, 256, 0);
  gemm_kernel<<<dim3(1, 1024), 256, 0, stream>>>(bufA, 128, Wp5, b5, bufB, 64, 64, 128, 0);

  final_layer_kernel<<<(2 * BATCH + 255) / 256, 256, 0, stream>>>(bufB, W6, b6, sres);
  loss_kernel<<<1, 1024, 0, stream>>>(sres, (float*)d_out);
}